// FreqDecoupledBlock_80530636800258
// MI455X (gfx1250) — compile-verified
//
#include <hip/hip_runtime.h>
#include <math.h>

// ---------------- problem constants (from reference) ----------------
#define BB 4
#define CC 64
#define HH 128
#define WW 128
#define LL (HH*WW)           // 16384
#define BL (BB*LL)           // 65536 rows of the flattened sequence
#define DI 128               // d_inner
#define DTR 4                // dt_rank
#define DSTATE 16
#define NXP 36               // dt_rank + 2*D_STATE
#define WB 65                // rfft bins along W
#define FR (BB*HH*WB)        // 33280 frequency rows (b,h,bin)
#define NF (FR*CC)           // 2,129,920 freq elements per plane

typedef float v2f __attribute__((ext_vector_type(2)));
typedef float v8f __attribute__((ext_vector_type(8)));

__device__ __forceinline__ float sigmoidf_(float v){ return 1.0f/(1.0f+__expf(-v)); }

// ================= pack x (B,C,H,W) -> xf (B*L, C), LDS tiled transpose =================
__global__ void k_pack_xf(const float* __restrict__ x, float* __restrict__ xf){
  __shared__ float tile[64][65];
  int b  = blockIdx.x >> 8;            // 256 pixel-blocks per batch (L/64)
  int p0 = (blockIdx.x & 255) << 6;
  int tp = threadIdx.x & 63;
  int t4 = threadIdx.x >> 6;
  #pragma unroll
  for (int cc=0; cc<64; cc+=4){
    int c = cc + t4;
    tile[c][tp] = x[((size_t)(b*CC + c))*LL + p0 + tp];   // coalesced over pixels
  }
  __syncthreads();
  #pragma unroll
  for (int pp=0; pp<64; pp+=4){
    int p = pp + t4;
    xf[((size_t)b*LL + p0 + p)*CC + tp] = tile[tp][p];    // coalesced over channels
  }
}

// ================= generic fp32 WMMA GEMM: OUT[M,N] = A[M,K] * B[K,N] (+bias, epilogue) ====
// EPI: 0=none 1=relu 2=softplus ; BT: B stored transposed as W[n*K+k] ; NCHW: scatter-store
// NT: N-tiles per wave (multi-accumulator, A loaded once per k-step, NT back-to-back WMMAs)
// GUARD: enable N-bounds checks (only needed when N is not a multiple of 16*NT coverage)
// A-operand ISA layout: lanes0-15 hold rows M=lane, K=k0..k0+1 ; lanes16-31 K=k0+2..k0+3.
template<int EPI, int BT, int NCHW, int NT, int GUARD>
__global__ void k_wmma_gemm(const float* __restrict__ A, int lda,
                            const float* __restrict__ Bm, int ldb,
                            const float* __restrict__ bias,
                            float* __restrict__ Out, int ldo,
                            int M, int N, int K, int gn, int tiles_total,
                            int nchwC, int nchwL)
{
  (void)M;
  int wave = threadIdx.x >> 5;
  int lane = threadIdx.x & 31;
  int tile = blockIdx.x * (blockDim.x >> 5) + wave;
  if (tile >= tiles_total) return;               // wave-uniform: keeps EXEC all-ones for WMMA
  int tm  = tile / gn;                           // M-tile
  int tng = tile - tm*gn;                        // N-tile group (covers NT*16 columns)

  int mrow  = (tm<<4) + (lane & 15);
  int khalf = (lane >> 4) << 1;                  // 0 or 2
  int ncol0 = tng*(NT*16) + (lane & 15);

  v8f acc[NT];
  #pragma unroll
  for (int nt=0; nt<NT; ++nt) acc[nt] = (v8f){0.f,0.f,0.f,0.f,0.f,0.f,0.f,0.f};

  for (int k0 = 0; k0 < K; k0 += 4){
    int kk = k0 + khalf;
    v2f a = *(const v2f*)(A + (size_t)mrow*lda + kk);     // A[mrow][kk], A[mrow][kk+1]
    #pragma unroll
    for (int nt=0; nt<NT; ++nt){
      int ncol = ncol0 + nt*16;
      v2f bv;
      if (GUARD && ncol >= N){
        bv = (v2f){0.f, 0.f};
      } else if (BT){
        bv = *(const v2f*)(Bm + (size_t)ncol*K + kk);     // contiguous pair
      } else {
        bv = (v2f){ Bm[(size_t)kk*ldb + ncol], Bm[(size_t)(kk+1)*ldb + ncol] };
      }
      // D = A(16x4,f32) x B(4x16,f32) + C   -> v_wmma_f32_16x16x4_f32
      acc[nt] = __builtin_amdgcn_wmma_f32_16x16x4_f32(
                  false, a, false, bv, (short)0, acc[nt], false, false);
    }
  }

  #pragma unroll
  for (int nt=0; nt<NT; ++nt){
    int ncol = ncol0 + nt*16;
    if (GUARD && ncol >= N) continue;
    float bb = (bias != nullptr) ? bias[ncol] : 0.f;
    #pragma unroll
    for (int r=0;r<8;++r){
      int row = (tm<<4) + r + ((lane>>4)<<3);    // D layout: vgpr r -> M=r (lo), M=r+8 (hi)
      float v = acc[nt][r] + bb;
      if (EPI==1) v = fmaxf(v, 0.f);
      else if (EPI==2) v = (v > 20.f) ? v : log1pf(__expf(v));  // softplus
      if (NCHW){
        int bidx = row / nchwL; int pix = row - bidx*nchwL;
        Out[((size_t)(bidx*nchwC + ncol))*nchwL + pix] = v;
      } else {
        Out[(size_t)row*ldo + ncol] = v;
      }
    }
  }
}

// ================= depthwise causal conv (k=4) + SiLU on x_in = xz[:, :128] =================
__global__ void k_conv_silu(const float* __restrict__ xz, const float* __restrict__ cw,
                            const float* __restrict__ cb, float* __restrict__ xc){
  size_t tid = (size_t)blockIdx.x*blockDim.x + threadIdx.x;   // BL*DI threads
  int d = (int)(tid & (DI-1));
  size_t i = tid >> 7;                 // row in [0, BL)
  int t = (int)(i & (LL-1));
  size_t brow = i - (size_t)t;         // b*L
  float acc = cb[d];
  #pragma unroll
  for (int k=0;k<4;++k){
    int tt = t - 3 + k;
    if (tt >= 0) acc += cw[d*4+k] * xz[(brow + (size_t)tt)*256 + d];
  }
  xc[i*DI + d] = acc * sigmoidf_(acc);
}

// ================= selective scan: 8192 lanes = (b, d, s); shfl reduction over 16 states ====
__global__ void k_scan(const float* __restrict__ xc, const float* __restrict__ dtb,
                       const float* __restrict__ dbc, const float* __restrict__ A_log,
                       float* __restrict__ y){
  int tid = blockIdx.x*blockDim.x + threadIdx.x;     // 8192 total
  int b   = tid >> 11;
  int rem = tid & 2047;
  int d   = rem >> 4;
  int s   = rem & 15;
  float Ads = -__expf(A_log[d*DSTATE + s]);          // A = -exp(A_log)
  float h = 0.f;
  size_t base = (size_t)b * LL;
  for (int t=0; t<LL; ++t){
    size_t i = base + t;
    float dtv = dtb[i*DI + d];
    float xv  = xc [i*DI + d];
    float Bv  = dbc[i*NXP + DTR + s];
    float Cv  = dbc[i*NXP + DTR + DSTATE + s];
    h = h * __expf(dtv*Ads) + dtv*xv*Bv;
    float p = h * Cv;
    p += __shfl_xor(p, 1);
    p += __shfl_xor(p, 2);
    p += __shfl_xor(p, 4);
    p += __shfl_xor(p, 8);
    if (s == 0) y[i*DI + d] = p;
  }
}

// ================= y = (scan + xc*D) * silu(z) =================
__global__ void k_gate(float* __restrict__ y, const float* __restrict__ xc,
                       const float* __restrict__ Dp, const float* __restrict__ xz){
  size_t tid = (size_t)blockIdx.x*blockDim.x + threadIdx.x;   // BL*DI
  int d = (int)(tid & (DI-1));
  size_t i = tid >> 7;
  float z = xz[i*256 + 128 + d];
  y[i*DI + d] = (y[i*DI + d] + xc[i*DI + d]*Dp[d]) * (z * sigmoidf_(z));
}

// ================= 128-point radix-2 FFT stages over LDS (one wave per transform) ==========
__device__ void fft128_stages(float* sre, float* sim, int lane, float sgn){
  for (int st=0; st<7; ++st){
    int half = 1<<st;
    __syncthreads();
    #pragma unroll
    for (int j=0;j<2;++j){
      int bf  = lane + (j<<5);           // 64 butterflies / stage, 2 per lane
      int grp = bf >> st;
      int pos = bf & (half-1);
      int i0  = (grp << (st+1)) + pos;
      int i1  = i0 + half;
      float ang = sgn * 3.14159265358979323846f * (float)pos / (float)half;
      float sw, cw; __sincosf(ang, &sw, &cw);
      float re1 = sre[i1], im1 = sim[i1];
      float tre = re1*cw - im1*sw;
      float tim = re1*sw + im1*cw;
      float re0 = sre[i0], im0 = sim[i0];
      sre[i0] = re0 + tre; sim[i0] = im0 + tim;
      sre[i1] = re0 - tre; sim[i1] = im0 - tim;
    }
  }
  __syncthreads();
}
__device__ __forceinline__ int brev7(int i){ return (int)(__brev((unsigned)i) >> 25); }

// forward real FFT along W for every (b,c,h) row -> 65 bins at (b,c,h,bin)
__global__ void k_fft_rows_fwd(const float* __restrict__ x,
                               float* __restrict__ rRe, float* __restrict__ rIm){
  __shared__ float sre[4][128], sim[4][128];
  int wave = threadIdx.x >> 5, lane = threadIdx.x & 31;
  int row = blockIdx.x*4 + wave;                 // row = (b*C+c)*H + h, 32768 exact
  const float* src = x + (size_t)row*WW;
  #pragma unroll
  for (int j=0;j<4;++j){
    int idx = lane + (j<<5);
    sre[wave][brev7(idx)] = src[idx];
    sim[wave][brev7(idx)] = 0.f;
  }
  fft128_stages(sre[wave], sim[wave], lane, -1.0f);
  #pragma unroll
  for (int j=0;j<3;++j){
    int k = lane + (j<<5);
    if (k < WB){
      rRe[(size_t)row*WB + k] = sre[wave][k];
      rIm[(size_t)row*WB + k] = sim[wave][k];
    }
  }
}

// forward complex FFT along H for every (b,c,bin); write transposed (b,h,bin,c)
__global__ void k_fft_cols_fwd(const float* __restrict__ rRe, const float* __restrict__ rIm,
                               float* __restrict__ Xre, float* __restrict__ Xim){
  __shared__ float sre[4][128], sim[4][128];
  int wave = threadIdx.x>>5, lane = threadIdx.x&31;
  int col = blockIdx.x*4 + wave;                 // 16640 exact
  int bin = col % WB; int bc = col / WB;         // bc = b*C + c
  #pragma unroll
  for (int j=0;j<4;++j){
    int h = lane + (j<<5);
    size_t ad = ((size_t)bc*HH + h)*WB + bin;
    sre[wave][brev7(h)] = rRe[ad];
    sim[wave][brev7(h)] = rIm[ad];
  }
  fft128_stages(sre[wave], sim[wave], lane, -1.0f);
  int c = bc & 63, b = bc >> 6;
  #pragma unroll
  for (int j=0;j<4;++j){
    int h = lane + (j<<5);
    size_t ad = (((size_t)b*HH + h)*WB + bin)*CC + c;
    Xre[ad] = sre[wave][h];
    Xim[ad] = sim[wave][h];
  }
}

__global__ void k_mag(const float* __restrict__ Xre, const float* __restrict__ Xim,
                      float* __restrict__ mag){
  size_t i = (size_t)blockIdx.x*blockDim.x + threadIdx.x;
  if (i < (size_t)NF){ float re=Xre[i], im=Xim[i]; mag[i] = sqrtf(re*re+im*im); }
}

// out = m2 * exp(i*phase) = (m2/|X|) * X  (|X|==0 -> m2 + 0i since angle(0)=0); in-place
__global__ void k_scale_phase(float* __restrict__ Xre, float* __restrict__ Xim,
                              const float* __restrict__ mag, const float* __restrict__ m2){
  size_t i = (size_t)blockIdx.x*blockDim.x + threadIdx.x;
  if (i < (size_t)NF){
    float m = mag[i], v = m2[i];
    if (m > 1e-30f){ float s = v/m; Xre[i]*=s; Xim[i]*=s; }
    else           { Xre[i] = v; Xim[i] = 0.f; }
  }
}

// inverse complex FFT along H for every (b,bin,c); scale 1/128; layout (b,h,bin,c) kept
__global__ void k_ifft_cols(const float* __restrict__ Xre, const float* __restrict__ Xim,
                            float* __restrict__ gRe, float* __restrict__ gIm){
  __shared__ float sre[4][128], sim[4][128];
  int wave = threadIdx.x>>5, lane = threadIdx.x&31;
  int col = blockIdx.x*4 + wave;                 // 16640 exact: (b, bin, c)
  int c = col & 63; int rest = col >> 6; int bin = rest % WB; int b = rest / WB;
  #pragma unroll
  for (int j=0;j<4;++j){
    int h = lane + (j<<5);
    size_t ad = (((size_t)b*HH + h)*WB + bin)*CC + c;
    sre[wave][brev7(h)] = Xre[ad];
    sim[wave][brev7(h)] = Xim[ad];
  }
  fft128_stages(sre[wave], sim[wave], lane, +1.0f);
  #pragma unroll
  for (int j=0;j<4;++j){
    int h = lane + (j<<5);
    size_t ad = (((size_t)b*HH + h)*WB + bin)*CC + c;
    gRe[ad] = sre[wave][h]*(1.0f/128.0f);
    gIm[ad] = sim[wave][h]*(1.0f/128.0f);
  }
}

// inverse real FFT along W (Hermitian expand 65->128) + fuse with spatial branch (in-place)
__global__ void k_ifft_rows_fuse(const float* __restrict__ gRe, const float* __restrict__ gIm,
                                 float* __restrict__ fused){
  __shared__ float sre[4][128], sim[4][128];
  int wave = threadIdx.x>>5, lane = threadIdx.x&31;
  int row = blockIdx.x*4 + wave;                 // 32768 exact: (b,h,c)
  int c = row & 63; int rest = row >> 6; int h = rest & 127; int b = rest >> 7;
  size_t base = (((size_t)b*HH + h)*WB)*CC + c;  // + k*CC
  #pragma unroll
  for (int j=0;j<2;++j){
    int k = lane + (j<<5);
    float re = gRe[base + (size_t)k*CC];
    float im = gIm[base + (size_t)k*CC];
    sre[wave][brev7(k)] = re;
    sim[wave][brev7(k)] = im;
    if (k > 0){
      int kk = 128 - k;
      sre[wave][brev7(kk)] = re;
      sim[wave][brev7(kk)] = -im;
    }
  }
  if (lane == 0){
    float re = gRe[base + (size_t)64*CC];
    float im = gIm[base + (size_t)64*CC];
    sre[wave][brev7(64)] = re;
    sim[wave][brev7(64)] = im;
  }
  fft128_stages(sre[wave], sim[wave], lane, +1.0f);
  size_t obase = ((size_t)b*LL + (size_t)h*WW)*CC + c;
  #pragma unroll
  for (int j=0;j<4;++j){
    int w = lane + (j<<5);
    size_t oa = obase + (size_t)w*CC;
    fused[oa] = fused[oa] + sre[wave][w]*(1.0f/128.0f);
  }
}

// ======================================================================================
extern "C" void kernel_launch(void* const* d_in, const int* in_sizes, int n_in,
                              void* d_out, int out_size, void* d_ws, size_t ws_size,
                              hipStream_t stream){
  (void)in_sizes; (void)n_in; (void)out_size; (void)ws_size;
  const float* x      = (const float*)d_in[0];
  const float* W_in   = (const float*)d_in[1];
  const float* conv_w = (const float*)d_in[2];
  const float* conv_b = (const float*)d_in[3];
  const float* W_xproj= (const float*)d_in[4];
  const float* W_dt   = (const float*)d_in[5];
  const float* b_dt   = (const float*)d_in[6];
  const float* A_log  = (const float*)d_in[7];
  const float* Dp     = (const float*)d_in[8];
  const float* W_out  = (const float*)d_in[9];
  const float* Wf1    = (const float*)d_in[10];
  const float* bf1    = (const float*)d_in[11];
  const float* Wf2    = (const float*)d_in[12];
  const float* bf2    = (const float*)d_in[13];
  const float* W_enh  = (const float*)d_in[14];
  const float* b_enh  = (const float*)d_in[15];
  const float* W_seg  = (const float*)d_in[16];
  const float* b_seg  = (const float*)d_in[17];

  // -------- workspace arena (floats) --------
  float* ws  = (float*)d_ws;
  float* xf  = ws;                        // BL*C        = 4,194,304
  float* xz  = xf  + (size_t)4194304;     // BL*256      = 16,777,216
  float* xc  = xz  + (size_t)16777216;    // BL*DI       = 8,388,608
  float* dbc = xc  + (size_t)8388608;     // BL*36       = 2,359,296
  float* dtb = dbc + (size_t)2359296;     // BL*DI       = 8,388,608
  float* yb  = dtb + (size_t)8388608;     // BL*DI       = 8,388,608
  float* xsp = yb  + (size_t)8388608;     // BL*C        = 4,194,304 (also "fused")
  // frequency-branch buffers overlay xz (dead after k_gate); 7 * 2,129,920 < 16,777,216
  float* rRe = xz;
  float* rIm = rRe + (size_t)NF;
  float* Xre = rIm + (size_t)NF;
  float* Xim = Xre + (size_t)NF;
  float* mg  = Xim + (size_t)NF;
  float* m1  = mg  + (size_t)NF;
  float* m2  = m1  + (size_t)NF;

  // ---------------- spatial (Mamba) branch ----------------
  k_pack_xf<<<1024, 256, 0, stream>>>(x, xf);

  { // xz = xf @ W_in : M=65536 N=256 K=64 ; 4 N-tiles/wave -> gn=4
    int gn = 4, tot = (BL/16)*gn;
    k_wmma_gemm<0,0,0,4,0><<<(tot+7)/8, 256, 0, stream>>>(xf,64, W_in,256, nullptr,
                                                          xz,256, BL,256,64, gn,tot, 0,0);
  }
  k_conv_silu<<<(BL*DI)/256, 256, 0, stream>>>(xz, conv_w, conv_b, xc);

  { // dbc = xc @ W_xproj : N=36 -> 3 guarded N-tiles in one group
    int gn = 1, tot = (BL/16)*gn;
    k_wmma_gemm<0,0,0,3,1><<<(tot+7)/8, 256, 0, stream>>>(xc,128, W_xproj,36, nullptr,
                                                          dbc,36, BL,36,128, gn,tot, 0,0);
  }
  { // dt = softplus(dbc[:, :4] @ W_dt + b_dt) : K=4, N=128 -> gn=2
    int gn = 2, tot = (BL/16)*gn;
    k_wmma_gemm<2,0,0,4,0><<<(tot+7)/8, 256, 0, stream>>>(dbc,36, W_dt,128, b_dt,
                                                          dtb,128, BL,128,4, gn,tot, 0,0);
  }
  k_scan<<<32, 256, 0, stream>>>(xc, dtb, dbc, A_log, yb);
  k_gate<<<(BL*DI)/256, 256, 0, stream>>>(yb, xc, Dp, xz);

  { // x_spatial = y @ W_out : M=65536 N=64 K=128 -> gn=1
    int gn = 1, tot = (BL/16)*gn;
    k_wmma_gemm<0,0,0,4,0><<<(tot+7)/8, 256, 0, stream>>>(yb,128, W_out,64, nullptr,
                                                          xsp,64, BL,64,128, gn,tot, 0,0);
  }

  // ---------------- frequency branch ----------------
  k_fft_rows_fwd<<<8192, 128, 0, stream>>>(x, rRe, rIm);
  k_fft_cols_fwd<<<4160, 128, 0, stream>>>(rRe, rIm, Xre, Xim);
  k_mag<<<NF/256, 256, 0, stream>>>(Xre, Xim, mg);

  { // m1 = relu(mag @ Wf1^T + bf1) ; m2 = m1 @ Wf2^T + bf2 : M=33280 N=64 K=64 -> gn=1
    int gn = 1, tot = (FR/16)*gn;
    k_wmma_gemm<1,1,0,4,0><<<(tot+7)/8, 256, 0, stream>>>(mg,64, Wf1,64, bf1,
                                                          m1,64, FR,64,64, gn,tot, 0,0);
    k_wmma_gemm<0,1,0,4,0><<<(tot+7)/8, 256, 0, stream>>>(m1,64, Wf2,64, bf2,
                                                          m2,64, FR,64,64, gn,tot, 0,0);
  }
  k_scale_phase<<<NF/256, 256, 0, stream>>>(Xre, Xim, mg, m2);
  k_ifft_cols<<<4160, 128, 0, stream>>>(Xre, Xim, rRe, rIm);
  k_ifft_rows_fuse<<<8192, 128, 0, stream>>>(rRe, rIm, xsp);   // fused in-place into xsp

  // ---------------- output heads (NCHW scatter-store into d_out) ----------------
  float* outp = (float*)d_out;
  { int gn = 1, tot = (BL/16)*gn;
    k_wmma_gemm<0,1,1,4,0><<<(tot+7)/8, 256, 0, stream>>>(xsp,64, W_enh,64, b_enh,
                                                          outp, 0, BL,64,64, gn,tot, CC, LL);
    k_wmma_gemm<0,1,1,4,0><<<(tot+7)/8, 256, 0, stream>>>(xsp,64, W_seg,64, b_seg,
                                                          outp + (size_t)BB*CC*LL, 0,
                                                          BL,64,64, gn,tot, CC, LL);
  }
}